// GeoGraph_9234179686473
// MI455X (gfx1250) — compile-verified
//
#include <hip/hip_runtime.h>
#include <math.h>

typedef __attribute__((ext_vector_type(2))) float v2f;
typedef __attribute__((ext_vector_type(8))) float v8f;
typedef __attribute__((ext_vector_type(4))) unsigned int v4u;
typedef __attribute__((ext_vector_type(4))) int v4i;
typedef __attribute__((ext_vector_type(8))) int v8i;

#define N_POI   50000
#define EMBED   128
#define N_EDGES 800000
#define NB      512
#define NL      64
#define E_AUG   (2 * N_EDGES + N_POI)

// fp32 WMMA: D = A(16x4) x B(4x16) + C, fp32 throughout (matches reference precision)
__device__ __forceinline__ v8f wmma_f32_16x16x4(v2f a, v2f b, v8f c) {
    return __builtin_amdgcn_wmma_f32_16x16x4_f32(false, a, false, b, (short)0, c, false, false);
}

// ---- Tensor Data Mover: DMA a 2D f32 tile (tileW x tileH) from row-major global
// (row stride = tensorW elements) into LDS at lds_byte_off, inserting 4 DWORDs of
// padding after every 128 DWORDs (=> 132-float LDS row stride, bank-conflict free).
__device__ __forceinline__ void tdm_load_tile_f32(const float* gptr, unsigned lds_byte_off,
                                                  unsigned tensorW, unsigned tensorH,
                                                  unsigned tileW, unsigned tileH) {
    unsigned long long ga = (unsigned long long)(size_t)gptr;
    v4u g0;
    g0.x = 1u;                                    // count=1, is_restore=0, gather off
    g0.y = lds_byte_off;                          // lds_addr
    g0.z = (unsigned)ga;                          // global_addr[31:0]
    g0.w = ((unsigned)(ga >> 32) & 0x01FFFFFFu) | (2u << 30);  // addr[56:32] | type=2
    v8i g1;
    g1.s0 = (int)((2u << 16)      // data_size = 4 bytes
                | (1u << 20)      // pad_enable
                | (6u << 22)      // pad_interval: 128 DWORDs
                | (3u << 25));    // pad_amount: 4 DWORDs
    g1.s1 = (int)(tensorW << 16);           // tensor_dim0[15:0] in [31:16]
    g1.s2 = (int)((tensorH & 0xFFFFu) << 16); // tensor_dim0 hi=0 | tensor_dim1[15:0]
    g1.s3 = (int)((tensorH >> 16) | (tileW << 16)); // tensor_dim1 hi | tile_dim0
    g1.s4 = (int)tileH;                     // tile_dim1 | tile_dim2=0
    g1.s5 = (int)tensorW;                   // tensor_dim0_stride[31:0]
    g1.s6 = (int)((tensorW * tensorH & 0xFFFFu) << 16); // stride0 hi=0 | stride1[15:0]
    g1.s7 = (int)((tensorW * tensorH) >> 16);           // stride1[47:16]
    v4i gz4 = { 0, 0, 0, 0 };
    v8i gz8 = { 0, 0, 0, 0, 0, 0, 0, 0 };
    __builtin_amdgcn_tensor_load_to_lds(g0, g1, gz4, gz4, gz8, 0);
}

// ---------------- graph preprocessing ----------------

__global__ void geo_deg_kernel(const int* __restrict__ e0, const int* __restrict__ e1,
                               float* __restrict__ deg) {
    int e = blockIdx.x * blockDim.x + threadIdx.x;
    if (e >= E_AUG) return;
    int src;
    if (e < N_EDGES)           src = e0[e];
    else if (e < 2 * N_EDGES)  src = e1[e - N_EDGES];
    else                       src = e - 2 * N_EDGES;
    unsafeAtomicAdd(deg + src, 1.0f);
}

__global__ void geo_edgew_kernel(const int* __restrict__ e0, const int* __restrict__ e1,
                                 const float* __restrict__ dvec, const float* __restrict__ deg,
                                 float* __restrict__ w) {
    int e = blockIdx.x * blockDim.x + threadIdx.x;
    if (e >= E_AUG) return;
    int src, dst; float dv;
    if (e < N_EDGES)          { src = e0[e];           dst = e1[e];           dv = dvec[e]; }
    else if (e < 2 * N_EDGES) { src = e1[e - N_EDGES]; dst = e0[e - N_EDGES]; dv = dvec[e - N_EDGES]; }
    else                      { src = dst = e - 2 * N_EDGES; dv = 0.0f; }
    w[e] = rsqrtf(deg[src] * deg[dst]) * expf(-dv * dv);
}

// ---------------- GNN layer: h = enc @ W^T + b (WMMA f32, TDM-staged LDS) ----------------
// one 16-row M tile per block; 8 waves each compute one 16-col N tile;
// W (128x128) and the A tile (16x128) are DMA'd into LDS by the Tensor Data Mover.

#define LDW 132  // 132 mod 64 == 4 -> conflict-free row access across 16 lanes

__global__ void __launch_bounds__(256)
gnn_gemm_kernel(const float* __restrict__ enc, const float* __restrict__ W,
                const float* __restrict__ bias, float* __restrict__ h) {
    extern __shared__ float sm[];
    float* sW = sm;                    // 128 x 132
    float* sA = sm + EMBED * LDW;      // 16  x 132
    const int mbase = blockIdx.x * 16;
    const int wave = threadIdx.x >> 5, lane = threadIdx.x & 31;
    if (wave == 0) {
        // DMA weights + activation tile into LDS; wave stalls only on tensorcnt
        tdm_load_tile_f32(W, 0u, EMBED, EMBED, EMBED, EMBED);
        tdm_load_tile_f32(enc + (size_t)mbase * EMBED, (unsigned)(EMBED * LDW * 4),
                          EMBED, N_POI, EMBED, 16);
        __builtin_amdgcn_s_wait_tensorcnt(0);
    }
    __syncthreads();
    const int colL = lane & 15, kb = (lane >> 4) * 2;
    const int ncol = wave * 16 + colL;
    v8f c = {};
    #pragma unroll 4
    for (int kk = 0; kk < EMBED; kk += 4) {
        v2f a, b;
        a.x = sA[colL * LDW + kk + kb];     a.y = sA[colL * LDW + kk + kb + 1];
        b.x = sW[ncol * LDW + kk + kb];     b.y = sW[ncol * LDW + kk + kb + 1];
        c = wmma_f32_16x16x4(a, b, c);
    }
    const float bv = bias[ncol];
    const int rbase = mbase + (lane >> 4) * 8;
    #pragma unroll
    for (int r = 0; r < 8; ++r)
        h[(size_t)(rbase + r) * EMBED + ncol] = c[r] + bv;
}

// ---------------- message scatter: agg[src] += w[e] * h[dst] ----------------
// 32 lanes per edge (float4 each), fp32 hardware atomics into L2-resident agg

__global__ void gnn_scatter_kernel(const float* __restrict__ h, const float* __restrict__ w,
                                   const int* __restrict__ e0, const int* __restrict__ e1,
                                   float* __restrict__ agg) {
    long long t = (long long)blockIdx.x * blockDim.x + threadIdx.x;
    if (t >= (long long)E_AUG * 32) return;
    int e = (int)(t >> 5);
    int c4 = (int)(t & 31) * 4;
    int src, dst;
    if (e < N_EDGES)          { src = e0[e];           dst = e1[e]; }
    else if (e < 2 * N_EDGES) { src = e1[e - N_EDGES]; dst = e0[e - N_EDGES]; }
    else                      { src = dst = e - 2 * N_EDGES; }
    const float we = w[e];
    float4 v = *(const float4*)(h + (size_t)dst * EMBED + c4);
    float* ap = agg + (size_t)src * EMBED + c4;
    unsafeAtomicAdd(ap + 0, we * v.x);
    unsafeAtomicAdd(ap + 1, we * v.y);
    unsafeAtomicAdd(ap + 2, we * v.z);
    unsafeAtomicAdd(ap + 3, we * v.w);
}

// ---------------- leaky-relu + L2 row normalize (one wave32 per row) ----------------

__global__ void gnn_norm_kernel(const float* __restrict__ agg, float* __restrict__ out) {
    int row = blockIdx.x * 8 + (threadIdx.x >> 5);
    if (row >= N_POI) return;
    int lane = threadIdx.x & 31;
    float4 v = *(const float4*)(agg + (size_t)row * EMBED + lane * 4);
    v.x = v.x >= 0.f ? v.x : 0.01f * v.x;
    v.y = v.y >= 0.f ? v.y : 0.01f * v.y;
    v.z = v.z >= 0.f ? v.z : 0.01f * v.z;
    v.w = v.w >= 0.f ? v.w : 0.01f * v.w;
    float s = v.x * v.x + v.y * v.y + v.z * v.z + v.w * v.w;
    #pragma unroll
    for (int off = 16; off > 0; off >>= 1) s += __shfl_xor(s, off, 32);
    float scale = 1.0f / fmaxf(sqrtf(s), 1e-12f);
    float4 o = { v.x * scale, v.y * scale, v.z * scale, v.w * scale };
    *(float4*)(out + (size_t)row * EMBED + lane * 4) = o;
}

// ---------------- fused MHA: one workgroup per session, all stages in LDS ----------------
// LDS: S 64x132 | QKV 64x388 | SC 64x68 | O 64x132 | MO 128  (~185 KB of 320 KB WGP LDS)

#define LQK 388   // >= 384, mod 64 == 4
#define LSC 68

__global__ void __launch_bounds__(256)
mha_kernel(const float* __restrict__ enc, const int* __restrict__ data_x,
           const float* __restrict__ inW, const float* __restrict__ inB,
           const float* __restrict__ outW, const float* __restrict__ outB,
           float* __restrict__ aggr) {
    extern __shared__ float sm[];
    float* S   = sm;                    // 64 x 132
    float* QKV = S + 64 * LDW;          // 64 x 388
    float* SC  = QKV + 64 * LQK;        // 64 x 68
    float* O   = SC + 64 * LSC;         // 64 x 132
    float* MO  = O + 64 * LDW;          // 128
    const int b = blockIdx.x;
    const int tid = threadIdx.x;
    const int wave = tid >> 5, lane = tid & 31;
    const int colL = lane & 15, kb = (lane >> 4) * 2;

    // warm in_proj_w into cache while gathering tokens (reused by all 512 blocks)
    for (int r = tid; r < 3 * EMBED; r += 256)
        __builtin_prefetch(inW + (size_t)r * EMBED, 0, 3);

    // gather session tokens
    for (int i = tid; i < 64 * 32; i += 256) {
        int l = i >> 5, c4 = (i & 31) * 4;
        int poi = data_x[b * NL + l];
        float4 v = *(const float4*)(enc + (size_t)poi * EMBED + c4);
        S[l * LDW + c4 + 0] = v.x; S[l * LDW + c4 + 1] = v.y;
        S[l * LDW + c4 + 2] = v.z; S[l * LDW + c4 + 3] = v.w;
    }
    __syncthreads();

    // qkv = S @ inW^T + inB : 4 M-tiles x 24 N-tiles, inW read from L2
    for (int t = wave; t < 96; t += 8) {
        int mt = t / 24, nt = t % 24;
        int jcol = nt * 16 + colL;
        v8f c = {};
        #pragma unroll 4
        for (int kk = 0; kk < EMBED; kk += 4) {
            v2f a, bf;
            a.x = S[(mt * 16 + colL) * LDW + kk + kb];
            a.y = S[(mt * 16 + colL) * LDW + kk + kb + 1];
            bf.x = inW[(size_t)jcol * EMBED + kk + kb];
            bf.y = inW[(size_t)jcol * EMBED + kk + kb + 1];
            c = wmma_f32_16x16x4(a, bf, c);
        }
        float bv = inB[jcol];
        int rbase = mt * 16 + (lane >> 4) * 8;
        #pragma unroll
        for (int r = 0; r < 8; ++r)
            QKV[(rbase + r) * LQK + jcol] = c[r] + bv;
    }
    __syncthreads();

    const float iscale = 0.1767766952966369f;  // 1/sqrt(32)
    for (int hh = 0; hh < 4; ++hh) {
        const int qo = hh * 32, ko = 128 + hh * 32, vo = 256 + hh * 32;
        // scores = Q_h @ K_h^T * iscale : 4x4 tiles
        for (int t = wave; t < 16; t += 8) {
            int mt = t >> 2, nt = t & 3;
            v8f c = {};
            #pragma unroll
            for (int kk = 0; kk < 32; kk += 4) {
                v2f a, bf;
                a.x = QKV[(mt * 16 + colL) * LQK + qo + kk + kb];
                a.y = QKV[(mt * 16 + colL) * LQK + qo + kk + kb + 1];
                bf.x = QKV[(nt * 16 + colL) * LQK + ko + kk + kb];      // B[k][n] = K[n][k]
                bf.y = QKV[(nt * 16 + colL) * LQK + ko + kk + kb + 1];
                c = wmma_f32_16x16x4(a, bf, c);
            }
            int rbase = mt * 16 + (lane >> 4) * 8;
            #pragma unroll
            for (int r = 0; r < 8; ++r)
                SC[(rbase + r) * LSC + nt * 16 + colL] = c[r] * iscale;
        }
        __syncthreads();
        // softmax per row (one wave / row, 2 cols per lane)
        for (int r = wave; r < 64; r += 8) {
            float x0 = SC[r * LSC + lane], x1 = SC[r * LSC + 32 + lane];
            float m = fmaxf(x0, x1);
            #pragma unroll
            for (int off = 16; off > 0; off >>= 1) m = fmaxf(m, __shfl_xor(m, off, 32));
            float e0 = __expf(x0 - m), e1 = __expf(x1 - m);
            float s = e0 + e1;
            #pragma unroll
            for (int off = 16; off > 0; off >>= 1) s += __shfl_xor(s, off, 32);
            float inv = 1.0f / s;
            SC[r * LSC + lane] = e0 * inv;
            SC[r * LSC + 32 + lane] = e1 * inv;
        }
        __syncthreads();
        // context O_h = P @ V_h : 4x2 tiles, one per wave
        {
            int mt = wave >> 1, nt = wave & 1;
            v8f c = {};
            #pragma unroll 4
            for (int kk = 0; kk < 64; kk += 4) {
                v2f a, bf;
                a.x = SC[(mt * 16 + colL) * LSC + kk + kb];
                a.y = SC[(mt * 16 + colL) * LSC + kk + kb + 1];
                bf.x = QKV[(kk + kb) * LQK + vo + nt * 16 + colL];
                bf.y = QKV[(kk + kb + 1) * LQK + vo + nt * 16 + colL];
                c = wmma_f32_16x16x4(a, bf, c);
            }
            int rbase = mt * 16 + (lane >> 4) * 8;
            #pragma unroll
            for (int r = 0; r < 8; ++r)
                O[(rbase + r) * LDW + hh * 32 + nt * 16 + colL] = c[r];
        }
        __syncthreads();
    }

    // mean over L (mean commutes with the linear out-proj), then out-proj
    if (tid < EMBED) {
        float s = 0.f;
        for (int l = 0; l < 64; ++l) s += O[l * LDW + tid];
        MO[tid] = s * (1.0f / 64.0f);
    }
    __syncthreads();
    if (tid < EMBED) {
        float s = outB[tid];
        for (int k = 0; k < EMBED; ++k) s += MO[k] * outW[(size_t)tid * EMBED + k];
        aggr[(size_t)b * EMBED + tid] = s;
    }
}

// ---------------- tar_embed gather ----------------

__global__ void tar_gather_kernel(const float* __restrict__ enc, const int* __restrict__ data_poi,
                                  float* __restrict__ out) {
    int t = blockIdx.x * blockDim.x + threadIdx.x;
    int bb = t >> 5;
    if (bb >= NB) return;
    int c4 = (t & 31) * 4;
    int poi = data_poi[bb];
    float4 v = *(const float4*)(enc + (size_t)poi * EMBED + c4);
    *(float4*)(out + (size_t)bb * EMBED + c4) = v;
}

// ---------------- launch ----------------

extern "C" void kernel_launch(void* const* d_in, const int* in_sizes, int n_in,
                              void* d_out, int out_size, void* d_ws, size_t ws_size,
                              hipStream_t stream) {
    const float* poi_embeds = (const float*)d_in[0];
    const int*   edges      = (const int*)d_in[1];   // [2, N_EDGES]
    const float* dist_vec   = (const float*)d_in[2];
    const int*   data_poi   = (const int*)d_in[3];
    const int*   data_x     = (const int*)d_in[4];
    const float* lin_w      = (const float*)d_in[5]; // [2,128,128]
    const float* lin_b      = (const float*)d_in[6]; // [2,128]
    const float* in_w       = (const float*)d_in[7]; // [384,128]
    const float* in_b       = (const float*)d_in[8];
    const float* out_w      = (const float*)d_in[9]; // [128,128]
    const float* out_b      = (const float*)d_in[10];
    float* out = (float*)d_out;

    const int* e0 = edges;
    const int* e1 = edges + N_EDGES;

    float* ws = (float*)d_ws;
    float* deg  = ws;                       // 50,048
    float* wEdg = ws + 50048;               // 1,650,048
    float* bufE = ws + 1700096;             // 6,400,000
    float* bufH = ws + 8100096;             // 6,400,000
    float* bufA = ws + 14500096;            // 6,400,000

    const size_t smGemm = (size_t)(EMBED * LDW + 16 * LDW) * 4;                       // ~74 KB
    const size_t smMha  = (size_t)(64 * LDW + 64 * LQK + 64 * LSC + 64 * LDW + 128) * 4; // ~185 KB

    (void)hipMemsetAsync(deg, 0, (size_t)N_POI * 4, stream);
    geo_deg_kernel<<<(E_AUG + 255) / 256, 256, 0, stream>>>(e0, e1, deg);
    geo_edgew_kernel<<<(E_AUG + 255) / 256, 256, 0, stream>>>(e0, e1, dist_vec, deg, wEdg);

    const float* enc_in = poi_embeds;
    float* layer_out[2] = { bufE, bufH };
    long long scatThreads = (long long)E_AUG * 32;
    int scatGrid = (int)((scatThreads + 255) / 256);

    for (int i = 0; i < 2; ++i) {
        (void)hipMemsetAsync(bufA, 0, (size_t)N_POI * EMBED * 4, stream);
        gnn_gemm_kernel<<<N_POI / 16, 256, smGemm, stream>>>(
            enc_in, lin_w + (size_t)i * EMBED * EMBED, lin_b + (size_t)i * EMBED, bufH);
        gnn_scatter_kernel<<<scatGrid, 256, 0, stream>>>(bufH, wEdg, e0, e1, bufA);
        gnn_norm_kernel<<<(N_POI + 7) / 8, 256, 0, stream>>>(bufA, layer_out[i]);
        enc_in = layer_out[i];
    }
    const float* enc_final = layer_out[1];

    mha_kernel<<<NB, 256, smMha, stream>>>(enc_final, data_x, in_w, in_b, out_w, out_b, out);
    tar_gather_kernel<<<(NB * 32 + 255) / 256, 256, 0, stream>>>(enc_final, data_poi,
                                                                 out + (size_t)NB * EMBED);
}